// MatchLoss_81269371174940
// MI455X (gfx1250) — compile-verified
//
#include <hip/hip_runtime.h>
#include <hip/hip_bf16.h>

#define NA 100000
#define NC 80
#define NO 2048
#define NB 128

typedef float v2f __attribute__((ext_vector_type(2)));
typedef float v8f __attribute__((ext_vector_type(8)));

// ---------------- Kernel 1: zero the touched scatter cells ----------------
__global__ void zero_cells_kernel(const int* __restrict__ matched_inds,
                                  const int* __restrict__ gt_labels,
                                  float* __restrict__ maxbuf) {
    int t = blockIdx.x * blockDim.x + threadIdx.x;   // t in [0, NO*NB)
    int o = t >> 7;          // / NB
    int lab = gt_labels[o] - 1;
    int anc = matched_inds[t];
    maxbuf[anc * NC + lab] = 0.0f;                   // racing identical writes: fine
}

// ---------------- Kernel 2: per-pair decode + IoU + row-max + scatter-max ----
__global__ void pair_kernel(const float* __restrict__ cls_scores,
                            const float* __restrict__ bbox_preds,
                            const float* __restrict__ gt_bboxes,
                            const int*   __restrict__ gt_labels,
                            const float* __restrict__ anchors,
                            const int*   __restrict__ matched_inds,
                            float* __restrict__ maxbuf,
                            float* __restrict__ clsp,
                            float* __restrict__ boxp,
                            float* __restrict__ positive_out) {
    __shared__ float sred[NB];
    const int o = blockIdx.x;
    const int b = threadIdx.x;
    const int idx = o * NB + b;
    const int i = matched_inds[idx];
    const int lab = gt_labels[o] - 1;

    // matched class prob
    float s = cls_scores[i * NC + lab];
    float cprob = 1.0f / (1.0f + __expf(-s));

    // decode predicted box
    float a0 = anchors[i * 4 + 0], a1 = anchors[i * 4 + 1];
    float a2 = anchors[i * 4 + 2], a3 = anchors[i * 4 + 3];
    float dx = bbox_preds[i * 4 + 0], dy = bbox_preds[i * 4 + 1];
    float dw = bbox_preds[i * 4 + 2], dh = bbox_preds[i * 4 + 3];
    float ax = (a0 + a2) * 0.5f, ay = (a1 + a3) * 0.5f;
    float aw = a2 - a0, ah = a3 - a1;
    float px = ax + dx * aw;
    float py = ay + dy * ah;
    float pw = aw * __expf(fminf(dw, 4.0f));
    float ph = ah * __expf(fminf(dh, 4.0f));
    float bx1 = px - pw * 0.5f, by1 = py - ph * 0.5f;
    float bx2 = px + pw * 0.5f, by2 = py + ph * 0.5f;

    // IoU vs gt box (broadcast load, hits cache)
    float g0 = gt_bboxes[o * 4 + 0], g1 = gt_bboxes[o * 4 + 1];
    float g2 = gt_bboxes[o * 4 + 2], g3 = gt_bboxes[o * 4 + 3];
    float ix1 = fmaxf(bx1, g0), iy1 = fmaxf(by1, g1);
    float ix2 = fminf(bx2, g2), iy2 = fminf(by2, g3);
    float inter  = fmaxf(ix2 - ix1, 0.0f) * fmaxf(iy2 - iy1, 0.0f);
    float area_p = fmaxf(bx2 - bx1, 0.0f) * fmaxf(by2 - by1, 0.0f);
    float area_g = (g2 - g0) * (g3 - g1);
    float iou = inter / fmaxf(area_p + area_g - inter, 1e-12f);

    float box_prob = fminf(fmaxf((iou - 0.3f) * 2.5f, 0.0f), 1.0f); // /(0.7-0.3)
    clsp[idx] = cprob;
    boxp[idx] = box_prob;
    positive_out[idx] = (iou >= 0.5f) ? 1.0f : 0.0f;

    // block row-max over B=128
    sred[b] = iou;
    __syncthreads();
    #pragma unroll
    for (int str = NB / 2; str > 0; str >>= 1) {
        if (b < str) sred[b] = fmaxf(sred[b], sred[b + str]);
        __syncthreads();
    }
    float rmax = sred[0];

    float obp = fminf(fmaxf(iou / fmaxf(rmax, 1e-12f), 0.0f), 1.0f);
    // scatter-max: non-negative floats order like their uint bit patterns
    atomicMax((unsigned int*)&maxbuf[i * NC + lab], __float_as_uint(obp));
}

// ---------------- Kernel 3: bag losses, WMMA f32 16x16x4 row-sums ----------
// One wave32 per 16 objects. A-matrix 16x4 layout: lanes 0-15 hold M=0..15
// with K=0,1 (vgpr 0,1); lanes 16-31 hold same M with K=2,3.  B = ones ->
// D[m,n] = row_sum(m). D layout: lane L(<16) holds D[r,L] for r=0..7 (objects
// 0..7), lane L(>=16) holds D[8+r, L-16] (objects 8..15).
__global__ void __launch_bounds__(32)
loss_kernel(const float* __restrict__ clsp,
            const float* __restrict__ boxp,
            const float* __restrict__ maxbuf,
            const int*   __restrict__ matched_inds,
            const int*   __restrict__ gt_labels,
            float* __restrict__ lossbuf) {
    const int o0   = blockIdx.x * 16;
    const int lane = threadIdx.x;
    const int m    = lane & 15;
    const int kb   = (lane >> 4) * 2;      // K-slot base: 0 or 2
    const int o    = o0 + m;
    const int lab  = gt_labels[o] - 1;

    float aw0 = 0.f, aw1 = 0.f, al0 = 0.f, al1 = 0.f;   // positive bag: sum w, sum w*l
    float nw0 = 0.f, nw1 = 0.f, nl0 = 0.f, nl1 = 0.f;   // negative bag

    #pragma unroll 2
    for (int k = 0; k < 2; ++k) {
        const int bstart = (kb + k) * 32;
        float sw = 0.f, swl = 0.f, snw = 0.f, snwl = 0.f;
        for (int j = 0; j < 32; ++j) {
            const int idx = o * NB + bstart + j;
            float cp = clsp[idx];
            float bp = boxp[idx];
            float lp = cp * bp;
            float wp = 1.0f / fmaxf(1.0f - lp, 1e-12f);
            int anc = matched_inds[idx];
            float np = 1.0f - maxbuf[anc * NC + lab];
            float ln = cp * np;
            float om = 1.0f - ln;
            float wn = ln / fmaxf(om * om, 1e-12f);
            sw += wp; swl += wp * lp; snw += wn; snwl += wn * ln;
        }
        if (k == 0) { aw0 = sw; al0 = swl; nw0 = snw; nl0 = snwl; }
        else        { aw1 = sw; al1 = swl; nw1 = snw; nl1 = snwl; }
    }

    v2f Aw  = {aw0, aw1};
    v2f Al  = {al0, al1};
    v2f An  = {nw0, nw1};
    v2f Anl = {nl0, nl1};
    v2f Bones = {1.0f, 1.0f};
    v8f cz = {0.f, 0.f, 0.f, 0.f, 0.f, 0.f, 0.f, 0.f};

    v8f dW  = __builtin_amdgcn_wmma_f32_16x16x4_f32(false, Aw,  false, Bones, (short)0, cz, false, false);
    v8f dWL = __builtin_amdgcn_wmma_f32_16x16x4_f32(false, Al,  false, Bones, (short)0, cz, false, false);
    v8f dN  = __builtin_amdgcn_wmma_f32_16x16x4_f32(false, An,  false, Bones, (short)0, cz, false, false);
    v8f dNL = __builtin_amdgcn_wmma_f32_16x16x4_f32(false, Anl, false, Bones, (short)0, cz, false, false);

    if (lane == 0 || lane == 16) {
        const int baseo = o0 + ((lane == 16) ? 8 : 0);
        #pragma unroll
        for (int r = 0; r < 8; ++r) {
            float sw = dW[r], swl = dWL[r], snw = dN[r], snwl = dNL[r];
            float rec = -fmaxf(logf(swl / sw), -100.0f);
            float prc = -fmaxf(logf(1.0f - snwl / snw), -100.0f);
            lossbuf[baseo + r] = rec + prc;
        }
    }
}

// ---------------- Kernel 4: deterministic final reduction ------------------
__global__ void reduce_kernel(const float* __restrict__ lossbuf,
                              float* __restrict__ out) {
    __shared__ float sred[256];
    const int t = threadIdx.x;
    float acc = 0.0f;
    #pragma unroll
    for (int j = t; j < NO; j += 256) acc += lossbuf[j];
    sred[t] = acc;
    __syncthreads();
    #pragma unroll
    for (int s = 128; s > 0; s >>= 1) {
        if (t < s) sred[t] += sred[t + s];
        __syncthreads();
    }
    if (t == 0) out[0] = sred[0] * (1.0f / (float)NO);
}

extern "C" void kernel_launch(void* const* d_in, const int* in_sizes, int n_in,
                              void* d_out, int out_size, void* d_ws, size_t ws_size,
                              hipStream_t stream) {
    const float* cls_scores   = (const float*)d_in[0];   // (A,C)
    const float* bbox_preds   = (const float*)d_in[1];   // (A,4)
    const float* gt_bboxes    = (const float*)d_in[2];   // (O,4)
    const int*   gt_labels    = (const int*)  d_in[3];   // (O,)
    const float* anchors      = (const float*)d_in[4];   // (A,4)
    const int*   matched_inds = (const int*)  d_in[5];   // (O,B)
    // d_in[6] = sparse_indices (redundant: derived from matched_inds/gt_labels)

    float* out = (float*)d_out;           // [0] = loss, [1..] = positive (O*B)

    char* ws = (char*)d_ws;
    float* maxbuf  = (float*)(ws);                              // A*C floats = 32,000,000 B
    float* clsp    = (float*)(ws + 32000000);                   // O*B floats
    float* boxp    = (float*)(ws + 33048576);                   // O*B floats
    float* lossbuf = (float*)(ws + 34097152);                   // O floats

    // 1) zero only the cells that will be read back
    zero_cells_kernel<<<(NO * NB) / 256, 256, 0, stream>>>(matched_inds, gt_labels, maxbuf);

    // 2) per-pair work + row-max + scatter-max
    pair_kernel<<<NO, NB, 0, stream>>>(cls_scores, bbox_preds, gt_bboxes, gt_labels,
                                       anchors, matched_inds, maxbuf, clsp, boxp,
                                       out + 1);

    // 3) bag losses (WMMA f32 row-sum reduction, 16 objects per wave)
    loss_kernel<<<NO / 16, 32, 0, stream>>>(clsp, boxp, maxbuf, matched_inds,
                                            gt_labels, lossbuf);

    // 4) deterministic scalar reduction
    reduce_kernel<<<1, 256, 0, stream>>>(lossbuf, out);
}